// TreeAttentionV3_52991306498321
// MI455X (gfx1250) — compile-verified
//
#include <hip/hip_runtime.h>

typedef float    v2f __attribute__((ext_vector_type(2)));
typedef float    v8f __attribute__((ext_vector_type(8)));
typedef uint32_t v4u __attribute__((ext_vector_type(4)));
typedef uint32_t v8u __attribute__((ext_vector_type(8)));

#define B_  4
#define N_  1024
#define NL_ 4
#define R_  256
#define C_  1024
#define H_  8
#define D_  32
#define T_THRESH 0.001f

__device__ __forceinline__ v8f wmma4(v2f a, v2f b, v8f c) {
    // V_WMMA_F32_16X16X4_F32 : D = A(16x4) * B(4x16) + C(16x16)
    return __builtin_amdgcn_wmma_f32_16x16x4_f32(
        false, a, false, b, (short)0, c, false, false);
}

// ---------------------------------------------------------------------------
// TDM: DMA a 2-D tile (tile_d0 x tile_d1, 4-byte elems, row pitch stride0)
// from global memory into LDS.  D# built in SGPRs (all args wave-uniform).
// Caller must be a single wave (scalar-branch gated) and must
// s_wait_tensorcnt before anyone reads the LDS tile.
// ---------------------------------------------------------------------------
__device__ __forceinline__ void tdm_load_2d(const void* gaddr, void* ldsPtr,
                                            uint32_t tensor_d0, uint32_t tensor_d1,
                                            uint32_t tile_d0, uint32_t tile_d1,
                                            uint32_t stride0)
{
    uint64_t ga  = (uint64_t)(uintptr_t)gaddr;
    uint32_t lds = (uint32_t)(uintptr_t)ldsPtr;   // low 32 bits == LDS offset
    v4u g0;
    g0[0] = 1u;                                            // count=1, no gather
    g0[1] = lds;                                           // lds_addr
    g0[2] = (uint32_t)ga;                                  // global_addr[31:0]
    g0[3] = (uint32_t)((ga >> 32) & 0x01FFFFFFu) | (2u << 30); // addr[56:32] | type=2
    v8u g1;
    g1[0] = 2u << 16;                                      // data_size=2 (4 bytes)
    g1[1] = (tensor_d0 & 0xFFFFu) << 16;                   // tensor_dim0[15:0]
    g1[2] = (tensor_d0 >> 16) | ((tensor_d1 & 0xFFFFu) << 16);
    g1[3] = (tensor_d1 >> 16) | (tile_d0 << 16);           // tile_dim0
    g1[4] = tile_d1 & 0xFFFFu;                             // tile_dim1 (tile_dim2=0)
    g1[5] = stride0;                                       // tensor_dim0_stride lo
    g1[6] = 0u;                                            // stride0 hi | stride1 lo
    g1[7] = 0u;
    asm volatile("tensor_load_to_lds %0, %1" :: "s"(g0), "s"(g1) : "memory");
}

// ---------------------------------------------------------------------------
// Kernel 1: qkv = x @ Wc + bc, scattered into q/k/v with layout [B][L][H][N][D]
// Block = 8 waves = 128 rows x 16 cols; B panel (Wc 256x16) TDM-staged in LDS.
// ---------------------------------------------------------------------------
__global__ void __launch_bounds__(256) qkv_kernel(
    const float* __restrict__ x, const float* __restrict__ Wc,
    const float* __restrict__ bc,
    float* __restrict__ qb, float* __restrict__ kb2, float* __restrict__ vb)
{
    __shared__ float ldsB[R_ * 16];                 // 16 KB: Wc[k][c0..c0+15]
    const int lane = threadIdx.x & 31;
    const int wid  = threadIdx.x >> 5;
    const int lo = lane & 15, hi2 = lane >> 4;

    int blk     = blockIdx.x;                       // 128 * 48 = 6144 blocks
    int rowBlk  = blk / 48;
    int colTile = blk % 48;
    int c0 = colTile << 4;
    int i0 = (rowBlk << 7) + (wid << 4);

    if (__builtin_amdgcn_readfirstlane(wid) == 0) { // one TDM issue per block
        tdm_load_2d(Wc + c0, ldsB, /*tensor_d0=*/768, /*tensor_d1=*/R_,
                    /*tile_d0=*/16, /*tile_d1=*/R_, /*stride0=*/768);
        __builtin_amdgcn_s_wait_tensorcnt(0);
    }
    __syncthreads();

    v8f acc = {};
    const float* arow = x + (size_t)(i0 + lo) * R_ + 2 * hi2;
    const float* bl   = ldsB + lo;
#pragma unroll 8
    for (int kb = 0; kb < R_; kb += 4) {
        v2f a = *(const v2f*)(arow + kb);           // A[lo][kb+2hi .. +1]
        v2f b;
        b.x = bl[(kb + 2 * hi2) * 16];              // B[kb+2hi][c0+lo] from LDS
        b.y = bl[(kb + 2 * hi2 + 1) * 16];
        acc = wmma4(a, b, acc);
    }

    int col   = c0 + lo;                            // 0..767
    float bias = bc[col];
    int which = col >> 8;                           // 0=q 1=k 2=v
    int rem   = col & 255;
    int hh    = rem >> 5;
    int dd    = rem & 31;
    float* dst = (which == 0) ? qb : (which == 1) ? kb2 : vb;
#pragma unroll
    for (int j = 0; j < 8; ++j) {
        int i  = i0 + j + 8 * hi2;
        int l  = i & 3;
        int bn = i >> 2;
        int n  = bn & (N_ - 1);
        int b  = bn >> 10;
        dst[(size_t)((((b * NL_ + l) * H_ + hh) * N_ + n)) * D_ + dd] = acc[j] + bias;
    }
}

// ---------------------------------------------------------------------------
// Kernel 2: per-(b,h) tree attention, flash-style, 16-row tile per wave.
// Pass 1: per-level row max / sumexp (online).  Pass 2: recompute scores,
// p_l = relu(softmax-T), comb = prod_l p_l, y_l += comb @ V_l.
// q/k/v layout [B][L][H][N][D].  y layout [B][N][C], c = (l*H+h)*D + d.
// ---------------------------------------------------------------------------
__global__ void __launch_bounds__(256) attn_kernel(
    const float* __restrict__ qb, const float* __restrict__ kb2,
    const float* __restrict__ vb, float* __restrict__ y)
{
    __shared__ float ldsbuf[8][16 * 17];            // per-wave transpose tile
    const int lane = threadIdx.x & 31;
    const int wid  = threadIdx.x >> 5;
    const int lo = lane & 15, hi2 = lane >> 4;

    int w  = blockIdx.x * 8 + wid;                  // 2048 waves
    int nt = w & 63;
    int bh = w >> 6;
    int h  = bh & 7, b = bh >> 3;
    int n0 = nt << 4;
    const float scale = 0.17677669529663687f;       // 1/sqrt(32)

    const float* Q[NL_]; const float* K[NL_]; const float* V[NL_];
#pragma unroll
    for (int l = 0; l < NL_; ++l) {
        size_t base = (size_t)((b * NL_ + l) * H_ + h) * N_ * D_;
        Q[l] = qb + base; K[l] = kb2 + base; V[l] = vb + base;
    }

    v2f aq[NL_][8];                                 // Q fragments, kept resident
#pragma unroll
    for (int l = 0; l < NL_; ++l)
#pragma unroll
        for (int kk = 0; kk < 8; ++kk)
            aq[l][kk] = *(const v2f*)(Q[l] + (size_t)(n0 + lo) * D_ + kk * 4 + 2 * hi2);

    float rmax[NL_][8], rsum[NL_][8];
#pragma unroll
    for (int l = 0; l < NL_; ++l)
#pragma unroll
        for (int j = 0; j < 8; ++j) { rmax[l][j] = -3.4e38f; rsum[l][j] = 0.0f; }

    // ---------------- pass 1: per-level softmax stats ----------------
    for (int m0 = 0; m0 < N_; m0 += 16) {
#pragma unroll
        for (int l = 0; l < NL_; ++l) {
            v8f s = {};
#pragma unroll
            for (int kk = 0; kk < 8; ++kk) {
                v2f bf = *(const v2f*)(K[l] + (size_t)(m0 + lo) * D_ + kk * 4 + 2 * hi2);
                s = wmma4(aq[l][kk], bf, s);
            }
#pragma unroll
            for (int j = 0; j < 8; ++j) {
                float xv = s[j] * scale;
                float t = xv;
                t = fmaxf(t, __shfl_xor(t, 1, 16));
                t = fmaxf(t, __shfl_xor(t, 2, 16));
                t = fmaxf(t, __shfl_xor(t, 4, 16));
                t = fmaxf(t, __shfl_xor(t, 8, 16));
                float mnew = fmaxf(rmax[l][j], t);
                float e = __expf(xv - mnew);
                e += __shfl_xor(e, 1, 16);
                e += __shfl_xor(e, 2, 16);
                e += __shfl_xor(e, 4, 16);
                e += __shfl_xor(e, 8, 16);
                rsum[l][j] = rsum[l][j] * __expf(rmax[l][j] - mnew) + e;
                rmax[l][j] = mnew;
            }
        }
    }
#pragma unroll
    for (int l = 0; l < NL_; ++l)
#pragma unroll
        for (int j = 0; j < 8; ++j) rsum[l][j] = 1.0f / rsum[l][j];

    // ------- pass 2: comb = prod_l relu(softmax-T); y_l += comb @ V_l -------
    v8f yacc[NL_][2];
#pragma unroll
    for (int l = 0; l < NL_; ++l) { yacc[l][0] = {}; yacc[l][1] = {}; }
    float* lp = &ldsbuf[wid][0];

    for (int m0 = 0; m0 < N_; m0 += 16) {
        float comb[8];
#pragma unroll
        for (int j = 0; j < 8; ++j) comb[j] = 1.0f;
#pragma unroll
        for (int l = 0; l < NL_; ++l) {
            v8f s = {};
#pragma unroll
            for (int kk = 0; kk < 8; ++kk) {
                v2f bf = *(const v2f*)(K[l] + (size_t)(m0 + lo) * D_ + kk * 4 + 2 * hi2);
                s = wmma4(aq[l][kk], bf, s);
            }
#pragma unroll
            for (int j = 0; j < 8; ++j) {
                float p = __expf(s[j] * scale - rmax[l][j]) * rsum[l][j] - T_THRESH;
                comb[j] *= fmaxf(p, 0.0f);
            }
        }
        // transpose comb (C-frag: row=j+8hi, col=lo) -> A-frags via LDS.
        // Same-wave DS ops are architecturally in-order; fences stop reordering.
        asm volatile("" ::: "memory");
#pragma unroll
        for (int j = 0; j < 8; ++j)
            lp[(j + 8 * hi2) * 17 + lo] = comb[j];
        asm volatile("" ::: "memory");
        v2f pa[4];
#pragma unroll
        for (int kb4 = 0; kb4 < 4; ++kb4) {
            pa[kb4].x = lp[lo * 17 + kb4 * 4 + 2 * hi2];
            pa[kb4].y = lp[lo * 17 + kb4 * 4 + 2 * hi2 + 1];
        }
        asm volatile("" ::: "memory");
#pragma unroll
        for (int l = 0; l < NL_; ++l) {
#pragma unroll
            for (int half = 0; half < 2; ++half) {
#pragma unroll
                for (int kb4 = 0; kb4 < 4; ++kb4) {
                    v2f bv;
                    const float* vp = V[l] + (size_t)(m0 + kb4 * 4 + 2 * hi2) * D_ + half * 16 + lo;
                    bv.x = vp[0];
                    bv.y = vp[D_];
                    yacc[l][half] = wmma4(pa[kb4], bv, yacc[l][half]);
                }
            }
        }
    }

#pragma unroll
    for (int l = 0; l < NL_; ++l)
#pragma unroll
        for (int half = 0; half < 2; ++half)
#pragma unroll
            for (int j = 0; j < 8; ++j) {
                int n = n0 + j + 8 * hi2;
                int c = (l * H_ + h) * D_ + half * 16 + lo;
                y[((size_t)b * N_ + n) * C_ + c] = yacc[l][half][j];
            }
}

// ---------------------------------------------------------------------------
// Kernel 3: out = y @ Wp + bp.  Block = 128 rows x 16 cols; Wp panel
// (1024 x 16 = 64 KB) TDM-staged in LDS, shared by all 8 waves.
// ---------------------------------------------------------------------------
__global__ void __launch_bounds__(256) proj_kernel(
    const float* __restrict__ y, const float* __restrict__ Wp,
    const float* __restrict__ bp, float* __restrict__ out)
{
    __shared__ float ldsB[C_ * 16];                 // 64 KB: Wp[k][c0..c0+15]
    const int lane = threadIdx.x & 31;
    const int wid  = threadIdx.x >> 5;
    const int lo = lane & 15, hi2 = lane >> 4;

    int blk     = blockIdx.x;                       // 32 * 64 = 2048 blocks
    int rowBlk  = blk >> 6;
    int colTile = blk & 63;
    int c0 = colTile << 4;
    int r0 = (rowBlk << 7) + (wid << 4);

    if (__builtin_amdgcn_readfirstlane(wid) == 0) {
        tdm_load_2d(Wp + c0, ldsB, /*tensor_d0=*/C_, /*tensor_d1=*/C_,
                    /*tile_d0=*/16, /*tile_d1=*/C_, /*stride0=*/C_);
        __builtin_amdgcn_s_wait_tensorcnt(0);
    }
    __syncthreads();

    v8f acc = {};
    const float* arow = y + (size_t)(r0 + lo) * C_ + 2 * hi2;
    const float* bl   = ldsB + lo;
#pragma unroll 8
    for (int kb = 0; kb < C_; kb += 4) {
        v2f a = *(const v2f*)(arow + kb);
        v2f b;
        b.x = bl[(kb + 2 * hi2) * 16];
        b.y = bl[(kb + 2 * hi2 + 1) * 16];
        acc = wmma4(a, b, acc);
    }
    float bias = bp[c0 + lo];
#pragma unroll
    for (int j = 0; j < 8; ++j)
        out[(size_t)(r0 + j + 8 * hi2) * C_ + c0 + lo] = acc[j] + bias;
}

// ---------------------------------------------------------------------------
extern "C" void kernel_launch(void* const* d_in, const int* in_sizes, int n_in,
                              void* d_out, int out_size, void* d_ws, size_t ws_size,
                              hipStream_t stream) {
    const float* x  = (const float*)d_in[0];
    const float* Wc = (const float*)d_in[1];
    const float* bc = (const float*)d_in[2];
    const float* Wp = (const float*)d_in[3];
    const float* bp = (const float*)d_in[4];
    float* out = (float*)d_out;

    float* ws = (float*)d_ws;
    const size_t QKV = (size_t)B_ * NL_ * H_ * N_ * D_;   // 4,194,304 floats
    float* q = ws;
    float* k = ws + QKV;
    float* v = ws + 2 * QKV;
    float* y = ws + 3 * QKV;                               // 64 MB total scratch

    qkv_kernel <<<6144, 256, 0, stream>>>(x, Wc, bc, q, k, v);
    attn_kernel<<< 256, 256, 0, stream>>>(q, k, v, y);
    proj_kernel<<<2048, 256, 0, stream>>>(y, Wp, bp, out);
}